// Codec_90890097918550
// MI455X (gfx1250) — compile-verified
//
#include <hip/hip_runtime.h>
#include <hip/hip_bf16.h>
#include <math.h>

#define RCH    3
#define IMW    128
#define IMH    128
#define NPLANE 24
#define KPIX   122              // 128 - 2*RCH
#define NSTEP  606              // wavefront f = kx + 4*ky in [0, 605]
#define STRD   50               // LDS row stride (floats): 8B aligned, bank-spread
#define LRELU(v) (fmaxf((v), 0.01f * (v)))

typedef float v8f __attribute__((ext_vector_type(8)));
typedef float v2f __attribute__((ext_vector_type(2)));
typedef unsigned int v4u __attribute__((ext_vector_type(4)));
typedef int v8i __attribute__((ext_vector_type(8)));
typedef int v4i __attribute__((ext_vector_type(4)));

struct __align__(16) Smem {
  float xpl[IMH * IMW];      // raw plane (TDM-loaded)
  float D[IMH * IMW];        // delta plane (zero-padded borders)
  float T[32 * STRD];        // feature matrix t (kept for residual)
  float A[32 * STRD];        // ping
  float B[32 * STRD];        // pong
  float WL1[48 * STRD];      // weights as W[n][k], rows padded
  float WL2[48 * STRD];
  float WL3[48 * STRD];
  float WL4[48 * STRD];
  float WL5[32 * STRD];      // rows 24..31 zero
  float WL6[16 * STRD];      // rows 12..15 zero
  float W7L[16];
  float biasL[7 * 48];       // b1..b7, 48-slot stride
  float ssq;
  unsigned hist[256];
};

// One 16x16 f32 output tile of In[32 x Kin] @ W^T (W stored as [n][k]).
// A layout (16x4 f32): lanes 0-15 M=lane K={0,1}; lanes 16-31 M=lane-16 K={2,3}.
// B layout (4x16 f32): lanes 0-15 N=lane K={0,1}; lanes 16-31 N=lane-16 K={2,3}.
template <int KSTEPS>
__device__ __forceinline__ v8f wmma_acc_tile(const float* In, const float* WT,
                                             int mt, int nt, int lane) {
  v8f c = {0.f, 0.f, 0.f, 0.f, 0.f, 0.f, 0.f, 0.f};
  const int klo = (lane & 16) ? 2 : 0;
  const float* aP = In + (mt * 16 + (lane & 15)) * STRD + klo;
  const float* bP = WT + (nt * 16 + (lane & 15)) * STRD + klo;
#pragma unroll
  for (int ks = 0; ks < KSTEPS; ++ks) {
    v2f a = *(const v2f*)(aP + 4 * ks);
    v2f b = *(const v2f*)(bP + 4 * ks);
    c = __builtin_amdgcn_wmma_f32_16x16x4_f32(false, a, false, b, (short)0, c,
                                              false, false);
  }
  return c;
}

// Full layer: Out = lrelu(In @ W^T + bias) [+ Res]; waves 0..2*NT-1 own tiles.
template <int NOUT, int KIN, bool RES>
__device__ __forceinline__ void mlp_layer(const float* In, float* Out,
                                          const float* WT, const float* bias,
                                          const float* Res, int wave, int lane) {
  constexpr int NT = (NOUT + 15) / 16;
  if (wave < 2 * NT) {
    const int mt = wave / NT, nt = wave % NT;
    v8f c = wmma_acc_tile<KIN / 4>(In, WT, mt, nt, lane);
    // D layout: VGPR r -> M = mt*16 + r + (lane>=16 ? 8 : 0), N = nt*16 + lane%16
    const int n = nt * 16 + (lane & 15);
    if (n < NOUT) {
      const float bv = bias[n];
      const int mbase = mt * 16 + ((lane & 16) ? 8 : 0);
#pragma unroll
      for (int r = 0; r < 8; ++r) {
        float v = LRELU(c[r] + bv);
        if (RES) v += Res[(mbase + r) * STRD + n];
        Out[(mbase + r) * STRD + n] = v;
      }
    }
  }
  __syncthreads();
}

__global__ __launch_bounds__(256) void codec_kernel(
    const float* __restrict__ x, const float* __restrict__ W1,
    const float* __restrict__ b1, const float* __restrict__ W2,
    const float* __restrict__ b2, const float* __restrict__ W3,
    const float* __restrict__ b3, const float* __restrict__ W4,
    const float* __restrict__ b4, const float* __restrict__ W5,
    const float* __restrict__ b5, const float* __restrict__ W6,
    const float* __restrict__ b6, const float* __restrict__ W7,
    const float* __restrict__ b7, float* __restrict__ wsSsq,
    unsigned* __restrict__ wsHist) {
  __shared__ Smem sm;
  const int tid = threadIdx.x;
  const int wave = tid >> 5;
  const int lane = tid & 31;
  const int p = blockIdx.x;
  const float* xsrc = x + (size_t)p * (IMH * IMW);

  // --- TDM: DMA the whole 128x128 f32 plane into LDS (one descriptor) ---
  if (wave == 0) {
    const unsigned long long ga = (unsigned long long)xsrc;
    const unsigned lds_off = (unsigned)(unsigned long long)(void*)&sm.xpl[0];
    v4u g0 = {0u, 0u, 0u, 0u};
    g0[0] = 1u;                                   // count = 1 valid descriptor
    g0[1] = lds_off;                              // LDS byte address
    g0[2] = (unsigned)(ga & 0xFFFFFFFFull);       // global_addr[31:0]
    g0[3] = (unsigned)((ga >> 32) & 0x01FFFFFFull) | (2u << 30);  // addr[56:32] | type=2
    v8i g1 = {0, 0, 0, 0, 0, 0, 0, 0};
    g1[0] = (int)(2u << 16);       // data_size = 4 bytes
    g1[1] = (int)(128u << 16);     // tensor_dim0[15:0] = 128 (bits 63:48)
    g1[2] = (int)(128u << 16);     // tensor_dim1[15:0] = 128 (bits 111:80 lo)
    g1[3] = (int)(128u << 16);     // tile_dim0 = 128 (bits 127:112)
    g1[4] = 128;                   // tile_dim1 = 128 (bits 143:128)
    g1[5] = 128;                   // tensor_dim0_stride = 128 (bits 191:160)
    const v4i z4 = {0, 0, 0, 0};
    const v8i z8 = {0, 0, 0, 0, 0, 0, 0, 0};
    __builtin_amdgcn_tensor_load_to_lds(g0, g1, z4, z4, z8, 0);
  }

  // --- zero delta plane, stats, and padded weight region ---
  for (int i = tid; i < IMH * IMW; i += 256) sm.D[i] = 0.f;
  {
    float* wz = sm.WL1;  // WL1..biasL are contiguous floats in the struct
    const int tot = 4 * 48 * STRD + 32 * STRD + 16 * STRD + 16 + 7 * 48;
    for (int i = tid; i < tot; i += 256) wz[i] = 0.f;
  }
  if (tid < 256) sm.hist[tid] = 0u;
  if (tid == 0) sm.ssq = 0.f;
  __syncthreads();  // zeros visible before fills

  // --- stage weights into LDS as W[n][k] with padded stride ---
  for (int i = tid; i < 48 * 48; i += 256) sm.WL1[(i / 48) * STRD + (i % 48)] = W1[i];
  for (int i = tid; i < 48 * 48; i += 256) sm.WL2[(i / 48) * STRD + (i % 48)] = W2[i];
  for (int i = tid; i < 48 * 48; i += 256) sm.WL3[(i / 48) * STRD + (i % 48)] = W3[i];
  for (int i = tid; i < 48 * 48; i += 256) sm.WL4[(i / 48) * STRD + (i % 48)] = W4[i];
  for (int i = tid; i < 24 * 48; i += 256) sm.WL5[(i / 48) * STRD + (i % 48)] = W5[i];
  for (int i = tid; i < 12 * 24; i += 256) sm.WL6[(i / 24) * STRD + (i % 24)] = W6[i];
  if (tid < 12) sm.W7L[tid] = W7[tid];
  if (tid < 48) sm.biasL[0 * 48 + tid] = b1[tid];
  if (tid < 48) sm.biasL[1 * 48 + tid] = b2[tid];
  if (tid < 48) sm.biasL[2 * 48 + tid] = b3[tid];
  if (tid < 48) sm.biasL[3 * 48 + tid] = b4[tid];
  if (tid < 24) sm.biasL[4 * 48 + tid] = b5[tid];
  if (tid < 12) sm.biasL[5 * 48 + tid] = b6[tid];
  if (tid < 1)  sm.biasL[6 * 48 + tid] = b7[tid];
  if (wave == 0) __builtin_amdgcn_s_wait_tensorcnt(0);
  __syncthreads();

  // --- 606 serial wavefront steps; each step batches <=31 pixels in M ---
  for (int f = 0; f < NSTEP; ++f) {
    const int kyMin = (f <= 121) ? 0 : ((f - 121 + 3) >> 2);
    int kyMax = f >> 2;
    if (kyMax > 121) kyMax = 121;
    const int nPix = kyMax - kyMin + 1;

    // Gather features: thread (m = tid&31, kg = tid>>5) fills T[m][kg*6 .. kg*6+5]
    {
      const int m = tid & 31, kg = tid >> 5;
      const int ky = kyMin + m;
      const bool valid = (m < nPix);
      const int yy = ky + RCH;
      const int xx = (f - 4 * ky) + RCH;
#pragma unroll
      for (int j = 0; j < 6; ++j) {
        const int k = kg * 6 + j;
        float v = 0.f;
        if (valid) {
          if (k < 21)       v = sm.xpl[(yy - 3 + k / 7) * IMW + (xx - 3 + k % 7)];
          else if (k < 24)  v = sm.xpl[yy * IMW + (xx - 3) + (k - 21)];
          else if (k < 45) { const int kk = k - 24;
                             v = sm.D[(yy - 3 + kk / 7) * IMW + (xx - 3 + kk % 7)]; }
          else              v = sm.D[yy * IMW + (xx - 3) + (k - 45)];
        }
        sm.T[m * STRD + k] = v;
      }
    }
    __syncthreads();

    mlp_layer<48, 48, false>(sm.T, sm.A, sm.WL1, sm.biasL + 0,   nullptr, wave, lane);
    mlp_layer<48, 48, false>(sm.A, sm.B, sm.WL2, sm.biasL + 48,  nullptr, wave, lane);
    mlp_layer<48, 48, false>(sm.B, sm.A, sm.WL3, sm.biasL + 96,  nullptr, wave, lane);
    mlp_layer<48, 48, true >(sm.A, sm.B, sm.WL4, sm.biasL + 144, sm.T,    wave, lane);
    mlp_layer<24, 48, false>(sm.B, sm.A, sm.WL5, sm.biasL + 192, nullptr, wave, lane);
    mlp_layer<12, 24, false>(sm.A, sm.B, sm.WL6, sm.biasL + 240, nullptr, wave, lane);

    // Layer 7 (12 -> 1) + delta + stats on wave 0 (one lane per pixel)
    if (wave == 0) {
      const int m = lane;
      float acc = sm.biasL[288];
#pragma unroll
      for (int k = 0; k < 12; ++k) acc += sm.B[m * STRD + k] * sm.W7L[k];
      const float pred = fminf(fmaxf(acc, -1.f), 1.f);
      float sq = 0.f;
      if (m < nPix) {
        const int ky = kyMin + m;
        const int yy = ky + RCH;
        const int xx = (f - 4 * ky) + RCH;
        const float delta = sm.xpl[yy * IMW + xx] - pred;
        sm.D[yy * IMW + xx] = delta;
        sq = delta * delta;
        if (delta >= -1.f && delta <= 1.f) {
          int bi = (int)floorf((delta + 1.f) * 128.f);
          if (bi > 255) bi = 255;
          atomicAdd(&sm.hist[bi], 1u);
        }
      }
#pragma unroll
      for (int off = 16; off > 0; off >>= 1) sq += __shfl_down(sq, off);
      if (lane == 0) sm.ssq += sq;   // single-writer deterministic accumulate
    }
    __syncthreads();
  }

  // Per-plane deterministic writeback (no cross-WG atomics)
  if (tid < 256) wsHist[p * 256 + tid] = sm.hist[tid];
  if (tid == 0) wsSsq[p] = sm.ssq;
}

__global__ __launch_bounds__(256) void finalize_kernel(
    const float* __restrict__ wsSsq, const unsigned* __restrict__ wsHist,
    float* __restrict__ out) {
  __shared__ float red[256];
  const int t = threadIdx.x;
  unsigned cnt = 0;
  for (int p = 0; p < NPLANE; ++p) cnt += wsHist[p * 256 + t];
  if (t == 128) cnt += NPLANE * KPIX;  // padded zero row 125 inside de-slice
  const float invN = 1.0f / 360144.0f;  // de.size = 8*3*123*122
  const float prob = (float)cnt * invN;
  red[t] = (prob > 0.f) ? (-prob * log2f(prob)) : 0.f;
  __syncthreads();
  for (int s = 128; s > 0; s >>= 1) {
    if (t < s) red[t] += red[t + s];
    __syncthreads();
  }
  if (t == 0) {
    float ssq = 0.f;
    for (int p = 0; p < NPLANE; ++p) ssq += wsSsq[p];
    out[0] = sqrtf(ssq / 387072.0f);  // deltas elems = 8*3*126*128
    out[1] = red[0] / 8.0f;
  }
}

extern "C" void kernel_launch(void* const* d_in, const int* in_sizes, int n_in,
                              void* d_out, int out_size, void* d_ws,
                              size_t ws_size, hipStream_t stream) {
  (void)in_sizes; (void)n_in; (void)out_size; (void)ws_size;
  const float* x  = (const float*)d_in[0];
  const float* W1 = (const float*)d_in[1];  const float* b1 = (const float*)d_in[2];
  const float* W2 = (const float*)d_in[3];  const float* b2 = (const float*)d_in[4];
  const float* W3 = (const float*)d_in[5];  const float* b3 = (const float*)d_in[6];
  const float* W4 = (const float*)d_in[7];  const float* b4 = (const float*)d_in[8];
  const float* W5 = (const float*)d_in[9];  const float* b5 = (const float*)d_in[10];
  const float* W6 = (const float*)d_in[11]; const float* b6 = (const float*)d_in[12];
  const float* W7 = (const float*)d_in[13]; const float* b7 = (const float*)d_in[14];
  float* wsSsq = (float*)d_ws;                                 // 24 floats
  unsigned* wsHist = (unsigned*)((char*)d_ws + 256);           // 24*256 uints
  codec_kernel<<<dim3(NPLANE), dim3(256), 0, stream>>>(
      x, W1, b1, W2, b2, W3, b3, W4, b4, W5, b5, W6, b6, W7, b7, wsSsq, wsHist);
  finalize_kernel<<<1, 256, 0, stream>>>(wsSsq, wsHist, (float*)d_out);
}